// CliffordProduct_85452669321821
// MI455X (gfx1250) — compile-verified
//
#include <hip/hip_runtime.h>

// Clifford (Cl(3,0)) geometric product: out(N,8) = flat(a ⊗ b)(N,64) @ M(64,8)
//
// Bandwidth-bound: 96 B / 128 flop per multivector (AI ~1.33) -> HBM roof at
// 23.3 TB/s is ~17 us for N=4Mi. Per wave-tile of 16 multivectors:
//  - VALU builds the 16x64 outer-product A-tile (dual-issued muls)
//  - 16x chained V_WMMA_F32_16X16X4_F32 contract K=64 against M on the matrix
//    pipe (co-executes with VALU; B fragments are loop-invariant)
//  - M's columns are REPLICATED into B cols 8..15 so D[m][n] = out[m][n&7]:
//    every lane stores 4 useful dwords from one base address, no EXEC masking
//    anywhere (WMMA requires EXEC all-1s), only a uniform scalar tail branch.
//  - a/b/out are touched exactly once -> nontemporal hints keep them out of L2.

typedef __attribute__((ext_vector_type(2))) float v2f;
typedef __attribute__((ext_vector_type(4))) float v4f;
typedef __attribute__((ext_vector_type(8))) float v8f;

__global__ __launch_bounds__(256) void clifford_wmma_kernel(
    const float* __restrict__ A, const float* __restrict__ B,
    const float* __restrict__ M, float* __restrict__ Out, int nRows)
{
    const int lane = threadIdx.x & 31;
    const int h    = lane >> 4;   // half-wave: K slots {0,1} vs {2,3}; D rows +8
    const int n    = lane & 15;   // B/D column; row-within-tile for loads
    const int col  = n & 7;       // logical output component
    const int hi   = n >> 3;      // 0: store D rows 0..3, 1: rows 4..7

    const int wavesPerBlock = blockDim.x >> 5;
    const int waveId = blockIdx.x * wavesPerBlock + (threadIdx.x >> 5);
    const int nWaves = gridDim.x * wavesPerBlock;

    // Loop-invariant B fragments: B[kk][n] = M[kk*8 + (n&7)] (cols replicated).
    // Layout (mirrors 32-bit A 16x4 table): vgpr v holds K = v + 2*h.
    v2f Bfrag[16];
#pragma unroll
    for (int c = 0; c < 16; ++c) {
        const int k0 = 4 * c + 2 * h;
        v2f t;
        t[0] = M[k0 * 8 + col];
        t[1] = M[(k0 + 1) * 8 + col];
        Bfrag[c] = t;
    }

    const int nTiles = (nRows + 15) >> 4;
    for (int tile = waveId; tile < nTiles; tile += nWaves) {
        const int base = tile << 4;
        const int row  = base + n;                      // lanes 16..31 mirror 0..15
        const int rc   = row < nRows ? row : nRows - 1; // clamp, never mask EXEC

        const v4f* ap = (const v4f*)(A + (size_t)rc * 8);
        const v4f* bp = (const v4f*)(B + (size_t)rc * 8);
        const v4f a01 = __builtin_nontemporal_load(ap);
        const v4f a23 = __builtin_nontemporal_load(ap + 1);
        const v4f b01 = __builtin_nontemporal_load(bp);
        const v4f b23 = __builtin_nontemporal_load(bp + 1);
        const float ar[8] = {a01[0], a01[1], a01[2], a01[3], a23[0], a23[1], a23[2], a23[3]};
        const float br[8] = {b01[0], b01[1], b01[2], b01[3], b23[0], b23[1], b23[2], b23[3]};

        v8f acc = {0.f, 0.f, 0.f, 0.f, 0.f, 0.f, 0.f, 0.f};
#pragma unroll
        for (int c = 0; c < 16; ++c) {
            // outer-product entries for this chunk's 4 K-slots
            // (compile-time indices -> registers only, no scratch)
            const float p0 = ar[(4 * c + 0) >> 3] * br[(4 * c + 0) & 7];
            const float p1 = ar[(4 * c + 1) >> 3] * br[(4 * c + 1) & 7];
            const float p2 = ar[(4 * c + 2) >> 3] * br[(4 * c + 2) & 7];
            const float p3 = ar[(4 * c + 3) >> 3] * br[(4 * c + 3) & 7];
            v2f Afrag;
            Afrag[0] = h ? p2 : p0;   // vgpr0: K = 2h
            Afrag[1] = h ? p3 : p1;   // vgpr1: K = 1 + 2h
            acc = __builtin_amdgcn_wmma_f32_16x16x4_f32(
                false, Afrag, false, Bfrag[c], (short)0, acc, false, false);
        }

        // Dense epilogue: lane stores D rows (8h + 4*hi + q), q=0..3, col n&7.
        const float s0 = hi ? acc[4] : acc[0];
        const float s1 = hi ? acc[5] : acc[1];
        const float s2 = hi ? acc[6] : acc[2];
        const float s3 = hi ? acc[7] : acc[3];
        const int rbase = base + 8 * h + 4 * hi;

        if (base + 16 <= nRows) {   // uniform branch (base is wave-uniform)
            float* p = Out + (size_t)rbase * 8 + col;
            __builtin_nontemporal_store(s0, p);
            __builtin_nontemporal_store(s1, p + 8);
            __builtin_nontemporal_store(s2, p + 16);
            __builtin_nontemporal_store(s3, p + 24);
        } else {
            // Rare tail tile: clamp rows. Clamped lanes loaded row nRows-1, so
            // they computed (and re-write) its exact value -- benign duplicates.
            const int r0 = rbase + 0 < nRows ? rbase + 0 : nRows - 1;
            const int r1 = rbase + 1 < nRows ? rbase + 1 : nRows - 1;
            const int r2 = rbase + 2 < nRows ? rbase + 2 : nRows - 1;
            const int r3 = rbase + 3 < nRows ? rbase + 3 : nRows - 1;
            Out[(size_t)r0 * 8 + col] = s0;
            Out[(size_t)r1 * 8 + col] = s1;
            Out[(size_t)r2 * 8 + col] = s2;
            Out[(size_t)r3 * 8 + col] = s3;
        }
    }
}

extern "C" void kernel_launch(void* const* d_in, const int* in_sizes, int n_in,
                              void* d_out, int out_size, void* d_ws, size_t ws_size,
                              hipStream_t stream) {
    const float* a = (const float*)d_in[0];
    const float* b = (const float*)d_in[1];
    const float* M = (const float*)d_in[2];
    float* out     = (float*)d_out;

    const int nRows = in_sizes[0] / 8;
    const int nTiles = (nRows + 15) / 16;
    const int wavesPerBlock = 256 / 32;
    const int tilesPerWave  = 8;  // grid-stride amortizes B-fragment setup
    int blocks = (nTiles + wavesPerBlock * tilesPerWave - 1) /
                 (wavesPerBlock * tilesPerWave);
    if (blocks < 1) blocks = 1;
    clifford_wmma_kernel<<<blocks, 256, 0, stream>>>(a, b, M, out, nRows);
}